// SSD_83236466196536
// MI455X (gfx1250) — compile-verified
//
#include <hip/hip_runtime.h>

// ---------------- problem constants (match reference) ----------------
#define B_IMG   64
#define N_ANCH  8732
#define N_CLS   21
#define N_CLS1  20          // classes 1..20
#define N_PRE   1000
#define N_PAD   1024        // padded top-k storage
#define TOPK_O  200
#define CONF_TH 0.01f
#define NMS_TH  0.5f
#define LBL_OFF 10000.0f
#define IMG_SZ  300.0f
#define NT      64          // 1024/16 tiles per dimension

typedef __attribute__((ext_vector_type(2))) float v2f;
typedef __attribute__((ext_vector_type(8))) float v8f;
typedef __attribute__((ext_vector_type(4))) int   v4i;

#define GAS __attribute__((address_space(1)))   // global
#define LAS __attribute__((address_space(3)))   // LDS

#if __has_builtin(__builtin_amdgcn_global_load_async_to_lds_b128) && \
    __has_builtin(__builtin_amdgcn_global_load_async_to_lds_b32)
#define HAVE_ASYNC_LDS 1
#else
#define HAVE_ASYNC_LDS 0
#endif

// ---------------- kernel 1: fused softmax + box decode ----------------
__global__ void k_decode_softmax(const float* __restrict__ loc,
                                 const float* __restrict__ conf,
                                 const float* __restrict__ anc,
                                 float* __restrict__ dec,
                                 float* __restrict__ probs)
{
    int idx = blockIdx.x * blockDim.x + threadIdx.x;
    if (idx >= B_IMG * N_ANCH) return;
    int a = idx % N_ANCH;
    const float* cf = conf + (size_t)idx * N_CLS;
    float m = cf[0];
#pragma unroll
    for (int c = 1; c < N_CLS; ++c) m = fmaxf(m, cf[c]);
    float ex[N_CLS];
    float s = 0.f;
#pragma unroll
    for (int c = 0; c < N_CLS; ++c) { ex[c] = __expf(cf[c] - m); s += ex[c]; }
    float inv = 1.f / s;
    float* pr = probs + (size_t)idx * N_CLS1;
#pragma unroll
    for (int c = 1; c < N_CLS; ++c) pr[c - 1] = ex[c] * inv;

    const float* l4 = loc + (size_t)idx * 4;
    const float* a4 = anc + (size_t)a * 4;
    float cx = a4[0] + l4[0] * 0.1f * a4[2];
    float cy = a4[1] + l4[1] * 0.1f * a4[3];
    float w  = a4[2] * __expf(l4[2] * 0.2f);
    float h  = a4[3] * __expf(l4[3] * 0.2f);
    float* d4 = dec + (size_t)idx * 4;
    d4[0] = (cx - 0.5f * w) * IMG_SZ;
    d4[1] = (cy - 0.5f * h) * IMG_SZ;
    d4[2] = (cx + 0.5f * w) * IMG_SZ;
    d4[3] = (cy + 0.5f * h) * IMG_SZ;
}

// ---------------- kernel 2: per-image top-1000 (histogram select + bitonic sort) ----------------
__global__ void k_topk(const float* __restrict__ probs,
                       const float* __restrict__ dec,
                       float* __restrict__ topS,
                       float* __restrict__ topB,
                       int*   __restrict__ topL)
{
    const int img  = blockIdx.x;
    const int tid  = threadIdx.x;
    const int NTHR = 256;
    const int TOT  = N_ANCH * N_CLS1;   // 174640

    __shared__ unsigned hist[2048];
    __shared__ unsigned long long keys[N_PAD];
    __shared__ unsigned s_tb, s_above, s_cA, s_cB;

    const float* p = probs + (size_t)img * TOT;

    for (int i = tid; i < 2048; i += NTHR) hist[i] = 0u;
    __syncthreads();
    // scores are softmax outputs in (0,1): positive floats -> bit pattern is order-preserving
    for (int i = tid; i < TOT; i += NTHR) {
        unsigned k = __float_as_uint(p[i]);
        atomicAdd(&hist[k >> 21], 1u);
    }
    __syncthreads();
    if (tid == 0) {
        unsigned cum = 0; int b = 2047;
        for (; b > 0; --b) { if (cum + hist[b] >= N_PRE) break; cum += hist[b]; }
        s_tb = (unsigned)b; s_above = cum; s_cA = 0u; s_cB = 0u;
    }
    __syncthreads();
    for (int i = tid; i < N_PAD; i += NTHR) keys[i] = 0ull;
    __syncthreads();
    unsigned tb = s_tb, above = s_above;
    for (int i = tid; i < TOT; i += NTHR) {
        unsigned k  = __float_as_uint(p[i]);
        unsigned bk = k >> 21;
        int slot = -1;
        if (bk > tb) slot = (int)atomicAdd(&s_cA, 1u);
        else if (bk == tb) {
            unsigned q = atomicAdd(&s_cB, 1u);
            if (above + q < N_PRE) slot = (int)(above + q);
        }
        if (slot >= 0)
            keys[slot] = ((unsigned long long)k << 32) | (unsigned)(~(unsigned)i);
    }
    __syncthreads();
    // bitonic sort, descending by (score bits, then smaller index first via ~i)
    for (unsigned sz = 2; sz <= N_PAD; sz <<= 1) {
        for (unsigned st = sz >> 1; st > 0; st >>= 1) {
            for (unsigned t = tid; t < N_PAD; t += NTHR) {
                unsigned j = t ^ st;
                if (j > t) {
                    unsigned long long x = keys[t], y = keys[j];
                    bool sw = ((t & sz) == 0) ? (x < y) : (x > y);
                    if (sw) { keys[t] = y; keys[j] = x; }
                }
            }
            __syncthreads();
        }
    }
    // gather sorted entries -> scores / boxes / labels
    for (int t = tid; t < N_PAD; t += NTHR) {
        unsigned long long key = keys[t];
        unsigned kb = (unsigned)(key >> 32);
        float sc = 0.f; int lbl = -1;
        float b0 = 0.f, b1 = 0.f, b2 = 0.f, b3 = 0.f;
        if (kb != 0u) {
            sc = __uint_as_float(kb);
            unsigned fi = ~(unsigned)key;
            int anchor = (int)(fi / N_CLS1);
            int cls    = (int)(fi % N_CLS1);
            lbl = cls + 1;
            const float* d4 = dec + ((size_t)img * N_ANCH + anchor) * 4;
            b0 = d4[0]; b1 = d4[1]; b2 = d4[2]; b3 = d4[3];
        }
        size_t o = (size_t)img * N_PAD + t;
        topS[o] = sc; topL[o] = lbl;
        topB[o * 4 + 0] = b0; topB[o * 4 + 1] = b1;
        topB[o * 4 + 2] = b2; topB[o * 4 + 3] = b3;
    }
}

// ---------------- kernel 3: zero suppression bitmask ----------------
__global__ void k_zero(unsigned* __restrict__ p, int n)
{
    int i = blockIdx.x * blockDim.x + threadIdx.x;
    if (i < n) p[i] = 0u;
}

// ---------------- kernel 4: 16x16-tiled IoU -> column bitmask ----------------
// Row-tile staged into LDS via GLOBAL_LOAD_ASYNC_TO_LDS (ASYNCcnt path);
// area sums for the whole 16x16 tile via one V_WMMA_F32_16X16X4_F32.
__global__ void k_ioumask(const float* __restrict__ topB,
                          const int*   __restrict__ topL,
                          unsigned*    __restrict__ mask)
{
    int tr = blockIdx.x & (NT - 1);
    int tc = blockIdx.x >> 6;
    if (tr > tc) return;                 // only r < c pairs can suppress
    int img  = blockIdx.y;
    int lane = threadIdx.x;              // 32 threads = 1 wave
    size_t base = (size_t)img * N_PAD;

    __shared__ __align__(16) float rbraw[16][4];   // raw (un-offset) row boxes
    __shared__ int rlbl[16];

    if (lane < 16) {
        size_t o = base + (size_t)(tr * 16 + lane);
#if HAVE_ASYNC_LDS
        __builtin_amdgcn_global_load_async_to_lds_b128(
            (GAS v4i*)(topB + o * 4),
            (LAS v4i*)&rbraw[lane][0], 0, 0);
        __builtin_amdgcn_global_load_async_to_lds_b32(
            (GAS int*)(topL + o),
            (LAS int*)&rlbl[lane], 0, 0);
#else
        rbraw[lane][0] = topB[o * 4 + 0];
        rbraw[lane][1] = topB[o * 4 + 1];
        rbraw[lane][2] = topB[o * 4 + 2];
        rbraw[lane][3] = topB[o * 4 + 3];
        rlbl[lane]     = topL[o];
#endif
    }
#if HAVE_ASYNC_LDS
#if __has_builtin(__builtin_amdgcn_s_wait_asynccnt)
    __builtin_amdgcn_s_wait_asynccnt(0);
#else
    asm volatile("s_wait_asynccnt 0x0" ::: "memory");
#endif
#endif
    __syncthreads();

    // column data (label offset applied in registers; offsets cancel in areas)
    int c = tc * 16 + (lane & 15);
    size_t oc = base + c;
    float coff = (float)topL[oc] * LBL_OFF;
    float cx1 = topB[oc * 4 + 0] + coff, cy1 = topB[oc * 4 + 1] + coff;
    float cx2 = topB[oc * 4 + 2] + coff, cy2 = topB[oc * 4 + 3] + coff;
    float carea = (cx2 - cx1) * (cy2 - cy1);

    // D[r][c] = area_r + area_c via rank-2 f32 WMMA:
    //   A(16x4) = [area_r, 1, 0, 0], B(4x16) = [1 ; area_c ; 0 ; 0]
    float rarea_l = 0.f;
    if (lane < 16)
        rarea_l = (rbraw[lane][2] - rbraw[lane][0]) *
                  (rbraw[lane][3] - rbraw[lane][1]);
#if __has_builtin(__builtin_amdgcn_wmma_f32_16x16x4_f32)
    v2f av, bv;
    av.x = (lane < 16) ? rarea_l : 0.f;       // A: lanes 0-15 hold K=0 / K=1
    av.y = (lane < 16) ? 1.f : 0.f;
    bv.x = (lane < 16) ? 1.f : 0.f;           // B: lanes 0-15 hold rows K=0 / K=1
    bv.y = (lane < 16) ? carea : 0.f;
    v8f dsum = {};
    dsum = __builtin_amdgcn_wmma_f32_16x16x4_f32(false, av, false, bv,
                                                 (short)0, dsum, false, false);
#else
    float dsum[8];
#pragma unroll
    for (int k = 0; k < 8; ++k) {
        int M = k + ((lane >> 4) << 3);
        dsum[k] = (rbraw[M][2] - rbraw[M][0]) * (rbraw[M][3] - rbraw[M][1]) + carea;
    }
#endif

    unsigned bits = 0u;
#pragma unroll
    for (int k = 0; k < 8; ++k) {
        int M  = k + ((lane >> 4) << 3);   // C/D layout: M = k + 8*(lane>=16), N = lane&15
        int rr = tr * 16 + M;
        float roff = (float)rlbl[M] * LBL_OFF;
        float rx1 = rbraw[M][0] + roff, ry1 = rbraw[M][1] + roff;
        float rx2 = rbraw[M][2] + roff, ry2 = rbraw[M][3] + roff;
        float iw = fminf(rx2, cx2) - fmaxf(rx1, cx1);
        float ih = fminf(ry2, cy2) - fmaxf(ry1, cy1);
        iw = fmaxf(iw, 0.f); ih = fmaxf(ih, 0.f);
        float inter = iw * ih;
        float su = dsum[k];
        // inter/(sum - inter + eps) > th  <=>  inter > th*(sum - inter + eps)
        bool hit = inter > NMS_TH * (su - inter + 1e-9f);
        if (hit && rr < c && rr < N_PRE && c < N_PRE) bits |= (1u << k);
    }
    if (bits) {
        unsigned word  = (unsigned)(tr >> 1);                          // (tr*16)/32
        unsigned shift = ((unsigned)(tr & 1) << 4) | ((unsigned)(lane >> 4) << 3);
        atomicOr(&mask[(base + c) * 32 + word], bits << shift);
    }
}

// ---------------- kernel 5: one-wave-per-image greedy NMS scan + emit top-200 ----------------
__global__ void k_nms_scan(const float* __restrict__ topS,
                           const float* __restrict__ topB,
                           const int*   __restrict__ topL,
                           const unsigned* __restrict__ mask,
                           float* __restrict__ boxes_o,
                           float* __restrict__ scores_o,
                           float* __restrict__ labels_o)
{
    int img  = blockIdx.x;
    int lane = threadIdx.x;              // 32 lanes; lane == bitmask word index
    size_t base = (size_t)img * N_PAD;

    unsigned keep_w = 0u;                // keep bits for indices [32*lane, 32*lane+31]
    for (int i = 0; i < N_PRE; ++i) {
        unsigned m = mask[(base + i) * 32 + lane];
        int sup = __any((int)(keep_w & m));
        bool kp = (topS[base + i] > CONF_TH) && !sup;
        if ((i >> 5) == lane) keep_w |= (kp ? 1u : 0u) << (i & 31);
    }

    __shared__ unsigned kw[32];
    __shared__ int list[TOPK_O];
    __shared__ int s_nk;
    kw[lane] = keep_w;
    __syncthreads();
    if (lane == 0) {
        int t = 0;
        for (int i = 0; i < N_PRE && t < TOPK_O; ++i)
            if ((kw[i >> 5] >> (i & 31)) & 1u) list[t++] = i;
        s_nk = t;
    }
    __syncthreads();
    int nk = s_nk;

    for (int t = lane; t < TOPK_O; t += 32) {
        float sc = 0.f, lb = -1.f;
        float b0 = 0.f, b1 = 0.f, b2 = 0.f, b3 = 0.f;
        if (t < nk) {
            size_t o = base + list[t];
            sc = topS[o]; lb = (float)topL[o];
            b0 = topB[o * 4 + 0]; b1 = topB[o * 4 + 1];
            b2 = topB[o * 4 + 2]; b3 = topB[o * 4 + 3];
        }
        size_t ob = (size_t)img * TOPK_O + t;
        boxes_o[ob * 4 + 0] = b0; boxes_o[ob * 4 + 1] = b1;
        boxes_o[ob * 4 + 2] = b2; boxes_o[ob * 4 + 3] = b3;
        scores_o[ob] = sc;
        labels_o[ob] = lb;
    }
}

// ---------------- launcher ----------------
extern "C" void kernel_launch(void* const* d_in, const int* in_sizes, int n_in,
                              void* d_out, int out_size, void* d_ws, size_t ws_size,
                              hipStream_t stream)
{
    (void)in_sizes; (void)n_in; (void)out_size; (void)ws_size;
    const float* loc  = (const float*)d_in[0];
    const float* conf = (const float*)d_in[1];
    const float* anc  = (const float*)d_in[2];

    char* ws = (char*)d_ws;
    size_t off = 0;
    auto carve = [&](size_t bytes) -> char* {
        char* p = ws + off;
        off += (bytes + 255) & ~(size_t)255;
        return p;
    };
    float*    dec   = (float*)   carve((size_t)B_IMG * N_ANCH * 4 * sizeof(float));
    float*    probs = (float*)   carve((size_t)B_IMG * N_ANCH * N_CLS1 * sizeof(float));
    float*    topS  = (float*)   carve((size_t)B_IMG * N_PAD * sizeof(float));
    float*    topB  = (float*)   carve((size_t)B_IMG * N_PAD * 4 * sizeof(float));
    int*      topL  = (int*)     carve((size_t)B_IMG * N_PAD * sizeof(int));
    unsigned* mask  = (unsigned*)carve((size_t)B_IMG * N_PAD * 32 * sizeof(unsigned));

    float* boxes_o  = (float*)d_out;                                   // (64,200,4)
    float* scores_o = boxes_o + (size_t)B_IMG * TOPK_O * 4;            // (64,200)
    float* labels_o = scores_o + (size_t)B_IMG * TOPK_O;               // (64,200)

    int n1 = B_IMG * N_ANCH;
    k_decode_softmax<<<(n1 + 255) / 256, 256, 0, stream>>>(loc, conf, anc, dec, probs);

    k_topk<<<B_IMG, 256, 0, stream>>>(probs, dec, topS, topB, topL);

    int nz = B_IMG * N_PAD * 32;
    k_zero<<<(nz + 255) / 256, 256, 0, stream>>>(mask, nz);

    dim3 g4(NT * NT, B_IMG);
    k_ioumask<<<g4, 32, 0, stream>>>(topB, topL, mask);

    k_nms_scan<<<B_IMG, 32, 0, stream>>>(topS, topB, topL, mask,
                                         boxes_o, scores_o, labels_o);
}